// InterBagAttention_86749749444743
// MI455X (gfx1250) — compile-verified
//
#include <hip/hip_runtime.h>
#include <hip/hip_bf16.h>

typedef __attribute__((ext_vector_type(2))) float v2f;
typedef __attribute__((ext_vector_type(8))) float v8f;
typedef int gvec4 __attribute__((vector_size(16)));                 // matches builtin pointee
typedef __attribute__((address_space(1))) gvec4* glob_v4i_ptr;      // global src
typedef __attribute__((address_space(3))) gvec4* lds_v4i_ptr;       // LDS dst

#define Gc 16
#define Bc 8
#define Lc 2048
#define Hc 768
#define ROWSTRIDE 772                    // 768 + 4 floats pad: de-conflicts LDS banks per row
#define NROWS (Bc + 1)                   // 8 data rows + 1 shared zero row for WMMA padding
#define WAVES_PER_BLOCK 2
#define TILE_FLOATS (NROWS * ROWSTRIDE)  // 6948
#define SIM_FLOATS 256                   // full 16x16 accumulator dump
#define WSLOT (TILE_FLOATS + SIM_FLOATS) // 7204 floats per wave

#ifndef __has_builtin
#define __has_builtin(x) 0
#endif

__global__ __launch_bounds__(WAVES_PER_BLOCK * 32)
void InterBagAttention_wmma_f32(const float* __restrict__ x,
                                float* __restrict__ out) {
  __shared__ float lds[WAVES_PER_BLOCK * WSLOT];
  const int tid  = threadIdx.x;
  const int wave = tid >> 5;
  const int lane = tid & 31;
  float* tile = &lds[wave * WSLOT];
  float* simb = tile + TILE_FLOATS;

  const int p = blockIdx.x * WAVES_PER_BLOCK + wave;  // (g,l) pair id
  const int g = p >> 11;                              // / Lc
  const int l = p & (Lc - 1);

  // ---- zero the shared padding row (row index Bc) ----
  #pragma unroll
  for (int i = 0; i < Hc / 128; ++i) {
    const int c = i * 128 + lane * 4;
    float4 z = {0.0f, 0.0f, 0.0f, 0.0f};
    *(float4*)(tile + Bc * ROWSTRIDE + c) = z;
  }

  // ---- cooperative coalesced fill of the 8x768 tile into LDS ----
  const size_t bagStride = (size_t)Lc * Hc;           // elements between bags
  const float* src = x + ((size_t)g * Bc) * bagStride + (size_t)l * Hc;
  #pragma unroll
  for (int b = 0; b < Bc; ++b) {
    const float* row = src + (size_t)b * bagStride;
    #pragma unroll
    for (int i = 0; i < Hc / 128; ++i) {              // 6 iters x 128 floats
      const int c = i * 128 + lane * 4;
#if __has_builtin(__builtin_amdgcn_global_load_async_to_lds_b128)
      __builtin_amdgcn_global_load_async_to_lds_b128(
          (glob_v4i_ptr)(row + c),
          (lds_v4i_ptr)(tile + b * ROWSTRIDE + c),
          /*offset=*/0, /*cpol=*/0);
#else
      const float4 v = *(const float4*)(row + c);
      *(float4*)(tile + b * ROWSTRIDE + c) = v;
#endif
    }
  }
  __builtin_amdgcn_wave_barrier();
  asm volatile("s_wait_asynccnt 0x0" ::: "memory");
  asm volatile("s_wait_dscnt 0x0" ::: "memory");

  // ---- sim = X * X^T via V_WMMA_F32_16X16X4_F32, K-loop over H ----
  // A fragment 16x4 f32: lane -> M = lane&15; VGPR0 holds K = k0+2*(lane>>4),
  // VGPR1 holds K+1. B fragment 4x16 (N striped across lanes) is identical to
  // the A fragment for the symmetric X*X^T product, so pass the same regs.
  // Pad lanes (M >= 8) read from the zero row -> branchless, EXEC stays full.
  const int half = lane >> 4;
  const int m    = lane & 15;
  const int r    = (m < Bc) ? m : Bc;                 // zero row for padding
  const float* abase = tile + r * ROWSTRIDE + 2 * half;

  v8f acc = {};
  for (int k0 = 0; k0 < Hc; k0 += 32) {               // 24 outer iterations
    v2f a[8];
    #pragma unroll
    for (int j = 0; j < 8; ++j)                       // batch 8 ds_load_b64
      a[j] = *(const v2f*)(abase + k0 + 4 * j);
    #pragma unroll
    for (int j = 0; j < 8; ++j)                       // then 8 back-to-back WMMAs
      acc = __builtin_amdgcn_wmma_f32_16x16x4_f32(false, a[j], false, a[j],
                                                  (short)0, acc, false, false);
  }

  // ---- dump 16x16 accumulator: acc[v] at lane L = sim[v + 8*(L>=16)][L&15]
  #pragma unroll
  for (int v = 0; v < 8; ++v)
    simb[(v + 8 * half) * 16 + m] = acc[v];
  __builtin_amdgcn_wave_barrier();
  asm volatile("s_wait_dscnt 0x0" ::: "memory");

  // ---- scores + softmax over 8 bags (redundant per lane; tiny) ----
  float w[Bc];
  float mx = -3.4e38f;
  #pragma unroll
  for (int b = 0; b < Bc; ++b) {
    float t = -1.0f;                       // subtract self-similarity constant
    #pragma unroll
    for (int c = 0; c < Bc; ++c) t += simb[b * 16 + c];
    w[b] = t;
    mx = fmaxf(mx, t);
  }
  float se = 0.0f;
  #pragma unroll
  for (int b = 0; b < Bc; ++b) { w[b] = __expf(w[b] - mx); se += w[b]; }
  const float inv = 1.0f / se;
  #pragma unroll
  for (int b = 0; b < Bc; ++b) w[b] *= inv;

  // ---- out[g][l][:] = sum_b w[b] * X[b][:], coalesced float4 stores ----
  float* dst = out + ((size_t)g * Lc + l) * Hc;
  #pragma unroll
  for (int i = 0; i < Hc / 128; ++i) {
    const int c = i * 128 + lane * 4;
    float4 o = {0.0f, 0.0f, 0.0f, 0.0f};
    #pragma unroll
    for (int b = 0; b < Bc; ++b) {
      const float4 vv = *(const float4*)(tile + b * ROWSTRIDE + c);
      o.x += w[b] * vv.x;
      o.y += w[b] * vv.y;
      o.z += w[b] * vv.z;
      o.w += w[b] * vv.w;
    }
    *(float4*)(dst + c) = o;
  }
}

extern "C" void kernel_launch(void* const* d_in, const int* in_sizes, int n_in,
                              void* d_out, int out_size, void* d_ws, size_t ws_size,
                              hipStream_t stream) {
  const float* x = (const float*)d_in[0];
  float* out = (float*)d_out;
  const int pairs = Gc * Lc;                       // 32768 (g,l) pairs
  dim3 grid(pairs / WAVES_PER_BLOCK);              // 16384 blocks
  dim3 block(WAVES_PER_BLOCK * 32);                // 64 threads = 2 waves
  hipLaunchKernelGGL(InterBagAttention_wmma_f32, grid, block, 0, stream, x, out);
}